// ASL_16638703305354
// MI455X (gfx1250) — compile-verified
//
#include <hip/hip_runtime.h>
#include <math.h>

typedef __attribute__((ext_vector_type(2))) float v2f;
typedef __attribute__((ext_vector_type(8))) float v8f;

#define BATCH   4
#define CHN     128
#define HW      4096      // 64*64
#define SEQL    4096
#define D_MODEL 64
#define D_INNER 128
#define D_STATE 16
#define DT_RANK 4
#define NPROJ   36        // DT_RANK + 2*D_STATE
#define NPAD    48
#define BL      16384     // BATCH * SEQL
#define MTOT    65536     // 4 mamba instances * BL
#define RTOT2   32768     // 2 branches * BL

#define EPI_NONE      0
#define EPI_SKIP      1
#define EPI_BIAS_WSEL 2
#define EPI_BIAS      3

__device__ __forceinline__ float sigmoidf_(float x) { return 1.0f / (1.0f + __expf(-x)); }
__device__ __forceinline__ float softplusf_(float x) {
    if (x > 20.0f) return x;
    return log1pf(__expf(x));
}

__device__ __forceinline__ v8f wmma4(float2 af, float2 bf, v8f c) {
    v2f a; a[0] = af.x; a[1] = af.y;
    v2f b; b[0] = bf.x; b[1] = bf.y;
    return __builtin_amdgcn_wmma_f32_16x16x4_f32(false, a, false, b, (short)0, c, false, false);
}

// ---------------------------------------------------------------------------
// Inner K-phase with one-stage software pipeline: next k-group's 5 fragments
// are loaded into live registers BEFORE the current group's 4 WMMAs issue, so
// waits are staggered (overlap of VMEM latency with the matrix pipe).
// ---------------------------------------------------------------------------
__device__ __forceinline__ void gemm_k_phase(const float* __restrict__ aPtr,
                                             size_t lda,
                                             const float* __restrict__ wPtr,
                                             int steps, v8f acc[4]) {
    if (steps <= 0) return;
    float2 bC  = *(const float2*)(wPtr);
    float2 a0C = *(const float2*)(aPtr);
    float2 a1C = *(const float2*)(aPtr + 16 * lda);
    float2 a2C = *(const float2*)(aPtr + 32 * lda);
    float2 a3C = *(const float2*)(aPtr + 48 * lda);
    for (int i = 0; i < steps - 1; ++i) {
        aPtr += 4;
        wPtr += 4;
        float2 bN  = *(const float2*)(wPtr);
        float2 a0N = *(const float2*)(aPtr);
        float2 a1N = *(const float2*)(aPtr + 16 * lda);
        float2 a2N = *(const float2*)(aPtr + 32 * lda);
        float2 a3N = *(const float2*)(aPtr + 48 * lda);
        acc[0] = wmma4(a0C, bC, acc[0]);
        acc[1] = wmma4(a1C, bC, acc[1]);
        acc[2] = wmma4(a2C, bC, acc[2]);
        acc[3] = wmma4(a3C, bC, acc[3]);
        bC = bN; a0C = a0N; a1C = a1N; a2C = a2N; a3C = a3N;
    }
    acc[0] = wmma4(a0C, bC, acc[0]);
    acc[1] = wmma4(a1C, bC, acc[1]);
    acc[2] = wmma4(a2C, bC, acc[2]);
    acc[3] = wmma4(a3C, bC, acc[3]);
}

// ---------------------------------------------------------------------------
// Generic fp32 WMMA GEMM: C[m,n] = sum_k A[m,k] * W[n,k]  (+ epilogue).
// Each wave computes a 64(M) x 16(N) tile. SPLIT: columns k >= kSplit come
// from A2 (column k-kSplit) — handled as a second contiguous K phase.
// ---------------------------------------------------------------------------
template <int MODE, bool SPLIT>
__global__ __launch_bounds__(256)
void gemm_wmma_f32(const float* __restrict__ A, const float* __restrict__ A2, int kSplit,
                   int M, int N, int K, int lda,
                   const float* __restrict__ W, int ldw,
                   float* __restrict__ C, int ldc,
                   const float* __restrict__ bias,
                   const float* __restrict__ sskip, const float* __restrict__ S, int ldS,
                   const float* __restrict__ w0p, const float* __restrict__ w1p, int rowSplit)
{
    int waveId = (blockIdx.x * blockDim.x + threadIdx.x) >> 5;
    int lane   = threadIdx.x & 31;
    int Nt = N >> 4;
    int Mt = M >> 6;
    if (waveId >= Mt * Nt) return;          // wave-uniform: EXEC stays all-1 for WMMA
    int nt = waveId % Nt;
    int mt = waveId / Nt;
    int nBase = nt << 4;
    int mBase = mt << 6;
    int half = lane >> 4;                   // which K-pair this lane carries
    int lr   = lane & 15;                   // A row / B col within fragment

    v8f acc[4];
    acc[0] = (v8f){}; acc[1] = (v8f){}; acc[2] = (v8f){}; acc[3] = (v8f){};

    const float* wPtr = W + (size_t)(nBase + lr) * ldw + 2 * half;

    if (!SPLIT) {
        const float* aPtr = A + (size_t)(mBase + lr) * lda + 2 * half;
        gemm_k_phase(aPtr, (size_t)lda, wPtr, K >> 2, acc);
    } else {
        const float* aPtr = A + (size_t)(mBase + lr) * lda + 2 * half;
        gemm_k_phase(aPtr, (size_t)lda, wPtr, kSplit >> 2, acc);
        const float* aPtr2 = A2 + (size_t)(mBase + lr) * lda + 2 * half;
        gemm_k_phase(aPtr2, (size_t)lda, wPtr + kSplit, (K - kSplit) >> 2, acc);
    }

    int col = nBase + lr;
    float ssv = 0.0f, bsv = 0.0f, w0v = 1.0f, w1v = 1.0f;
    if (MODE == EPI_SKIP) ssv = sskip[0];
    if (MODE == EPI_BIAS || MODE == EPI_BIAS_WSEL) bsv = bias[col];
    if (MODE == EPI_BIAS_WSEL) { w0v = w0p[0]; w1v = w1p[0]; }

#pragma unroll
    for (int s = 0; s < 4; ++s) {
#pragma unroll
        for (int r = 0; r < 8; ++r) {
            int row = mBase + s * 16 + half * 8 + r;
            float v = acc[s][r];
            if (MODE == EPI_SKIP)           v += ssv * S[(size_t)row * ldS + col];
            else if (MODE == EPI_BIAS_WSEL) v  = (v + bsv) * ((row < rowSplit) ? w0v : w1v);
            else if (MODE == EPI_BIAS)      v += bsv;
            C[(size_t)row * ldc + col] = v;
        }
    }
}

// ---------------------------------------------------------------------------
// Zero-pad x_proj_w (36 x 128) into WP (48 x 128) so the GEMM needs no guards.
// ---------------------------------------------------------------------------
__global__ void k_pad_xproj(const float* __restrict__ w, float* __restrict__ wp) {
    int i = blockIdx.x * blockDim.x + threadIdx.x;
    if (i >= NPAD * D_INNER) return;
    int r = i / D_INNER;
    wp[i] = (r < NPROJ) ? w[(size_t)r * D_INNER + (i % D_INNER)] : 0.0f;
}

// ---------------------------------------------------------------------------
// LayerNorm over 128 channels of (B,C,H,W) input, split into two 64-ch halves,
// written as 4 stacked mamba-instance matrices XH[(inst*BL + bl), 64].
// ---------------------------------------------------------------------------
__global__ void k_ln_split(const float* __restrict__ rgb, const float* __restrict__ ir,
                           const float* __restrict__ lnw, const float* __restrict__ lnb,
                           float* __restrict__ XH) {
    int row = blockIdx.x * blockDim.x + threadIdx.x;
    if (row >= RTOT2) return;
    int branch = row >> 14;
    int bl = row & (BL - 1);
    const float* src = (branch ? ir : rgb) + (size_t)(bl >> 12) * (CHN * HW) + (bl & (HW - 1));
    float s = 0.f;
    for (int c = 0; c < CHN; ++c) s += src[(size_t)c * HW];
    float mu = s * (1.0f / CHN);
    float v = 0.f;
    for (int c = 0; c < CHN; ++c) { float d = src[(size_t)c * HW] - mu; v += d * d; }
    float inv = rsqrtf(v * (1.0f / CHN) + 1e-5f);
    for (int c = 0; c < CHN; ++c) {
        float xn = (src[(size_t)c * HW] - mu) * inv * lnw[c] + lnb[c];
        int inst = branch * 2 + (c >> 6);
        XH[((size_t)inst * BL + bl) * D_MODEL + (c & 63)] = xn;
    }
}

// ---------------------------------------------------------------------------
// Depthwise causal conv (k=4) + SiLU on XR -> XC.
// ---------------------------------------------------------------------------
__global__ void k_conv_silu(const float* __restrict__ XR, const float* __restrict__ cw,
                            const float* __restrict__ cb, float* __restrict__ XC) {
    long idx = (long)blockIdx.x * blockDim.x + threadIdx.x;
    if (idx >= (long)MTOT * D_INNER) return;
    int d = (int)(idx & (D_INNER - 1));
    long m = idx >> 7;
    int l = (int)(m & (SEQL - 1));
    float acc = cb[d];
#pragma unroll
    for (int j = 0; j < 4; ++j) {
        int off = 3 - j;
        if (l >= off) acc += cw[d * 4 + j] * XR[(m - off) * D_INNER + d];
    }
    XC[idx] = acc * sigmoidf_(acc);
}

// ---------------------------------------------------------------------------
// Selective-scan: one block = one (instance,batch), one thread = one channel.
// Fuses dt_proj+softplus, SSM recurrence (16 states in registers), D-skip and
// SiLU(z) gating. dt/B/C row staged in LDS once per timestep.
// ---------------------------------------------------------------------------
__global__ __launch_bounds__(128)
void k_scan(const float* __restrict__ XDBL, const float* __restrict__ XC,
            const float* __restrict__ Z,
            const float* __restrict__ dtw, const float* __restrict__ dtb,
            const float* __restrict__ Alog, const float* __restrict__ Dskip,
            float* __restrict__ YG) {
    int ib = blockIdx.x;             // inst*BATCH + b, 0..15
    int d = threadIdx.x;             // 0..127
    __shared__ float sh[NPROJ];
    float w0 = dtw[d * 4 + 0], w1 = dtw[d * 4 + 1], w2 = dtw[d * 4 + 2], w3 = dtw[d * 4 + 3];
    float bdt = dtb[d];
    float Dd = Dskip[d];
    float Av[D_STATE], h[D_STATE];
#pragma unroll
    for (int s = 0; s < D_STATE; ++s) { Av[s] = -__expf(Alog[d * D_STATE + s]); h[s] = 0.f; }
    size_t base = (size_t)ib * SEQL;
    for (int t = 0; t < SEQL; ++t) {
        size_t row = base + t;
        if (d < NPROJ) sh[d] = XDBL[row * NPAD + d];
        __syncthreads();
        float dtv = sh[0] * w0 + sh[1] * w1 + sh[2] * w2 + sh[3] * w3 + bdt;
        float delta = softplusf_(dtv);
        float u  = XC[row * D_INNER + d];
        float zz = Z[row * D_INNER + d];
        float du = delta * u;
        float y = 0.f;
#pragma unroll
        for (int s = 0; s < D_STATE; ++s) {
            float e = __expf(delta * Av[s]);
            h[s] = h[s] * e + du * sh[DT_RANK + s];
            y += h[s] * sh[DT_RANK + D_STATE + s];
        }
        float g = zz * sigmoidf_(zz);
        YG[row * D_INNER + d] = (y + Dd * u) * g;
        __syncthreads();
    }
}

// ---------------------------------------------------------------------------
// Second LayerNorm over concat(m1, m2) per branch -> XM[(branch*BL+bl), 128].
// ---------------------------------------------------------------------------
__global__ void k_ln2(const float* __restrict__ Mb, const float* __restrict__ lnw,
                      const float* __restrict__ lnb, float* __restrict__ XM) {
    int row = blockIdx.x * blockDim.x + threadIdx.x;
    if (row >= RTOT2) return;
    int branch = row >> 14;
    int bl = row & (BL - 1);
    const float* m0 = Mb + ((size_t)(branch * 2 + 0) * BL + bl) * D_MODEL;
    const float* m1 = Mb + ((size_t)(branch * 2 + 1) * BL + bl) * D_MODEL;
    float s = 0.f;
    for (int c = 0; c < D_MODEL; ++c) s += m0[c] + m1[c];
    float mu = s * (1.0f / CHN);
    float v = 0.f;
    for (int c = 0; c < D_MODEL; ++c) {
        float d0 = m0[c] - mu; float d1 = m1[c] - mu;
        v += d0 * d0 + d1 * d1;
    }
    float inv = rsqrtf(v * (1.0f / CHN) + 1e-5f);
    float* dst = XM + (size_t)row * CHN;
    for (int c = 0; c < D_MODEL; ++c) {
        dst[c]           = (m0[c] - mu) * inv * lnw[c]           + lnb[c];
        dst[c + D_MODEL] = (m1[c] - mu) * inv * lnw[c + D_MODEL] + lnb[c + D_MODEL];
    }
}

// ---------------------------------------------------------------------------
// (b,l,o) -> (b,o,h,w) transpose of the final result.
// ---------------------------------------------------------------------------
__global__ void k_transpose_out(const float* __restrict__ TMP, float* __restrict__ out) {
    int idx = blockIdx.x * blockDim.x + threadIdx.x;
    if (idx >= BATCH * CHN * HW) return;
    int b  = idx >> 19;
    int ch = (idx >> 12) & (CHN - 1);
    int l  = idx & (HW - 1);
    out[idx] = TMP[((size_t)b * HW + l) * CHN + ch];
}

// ---------------------------------------------------------------------------
template <int MODE, bool SPLIT>
static void launch_gemm(hipStream_t st, const float* A, const float* A2, int kSplit,
                        int M, int N, int K, int lda, const float* W, int ldw,
                        float* C, int ldc, const float* bias,
                        const float* ss, const float* S, int ldS,
                        const float* w0, const float* w1, int rowSplit) {
    int waves = (M / 64) * (N / 16);
    int blocks = (waves + 7) / 8;
    gemm_wmma_f32<MODE, SPLIT><<<blocks, 256, 0, st>>>(A, A2, kSplit, M, N, K, lda, W, ldw,
                                                       C, ldc, bias, ss, S, ldS, w0, w1, rowSplit);
}

extern "C" void kernel_launch(void* const* d_in, const int* in_sizes, int n_in,
                              void* d_out, int out_size, void* d_ws, size_t ws_size,
                              hipStream_t stream) {
    const float* rgb        = (const float*)d_in[0];
    const float* ir         = (const float*)d_in[1];
    const float* ln_w       = (const float*)d_in[2];
    const float* ln_b       = (const float*)d_in[3];
    const float* in_proj_w  = (const float*)d_in[4];
    const float* conv_w     = (const float*)d_in[5];
    const float* conv_b     = (const float*)d_in[6];
    const float* x_proj_w   = (const float*)d_in[7];
    const float* dt_proj_w  = (const float*)d_in[8];
    const float* dt_proj_b  = (const float*)d_in[9];
    const float* A_log      = (const float*)d_in[10];
    const float* D_skip     = (const float*)d_in[11];
    const float* out_proj_w = (const float*)d_in[12];
    const float* proj_w     = (const float*)d_in[13];
    const float* proj_b     = (const float*)d_in[14];
    const float* skip_scale = (const float*)d_in[15];
    const float* weight_rgb = (const float*)d_in[16];
    const float* weight_ir  = (const float*)d_in[17];
    const float* convtout_w = (const float*)d_in[18];
    const float* convtout_b = (const float*)d_in[19];
    float* out = (float*)d_out;
    float* ws  = (float*)d_ws;

    // Workspace layout (floats). Later stages alias buffers that are dead.
    float* XH   = ws;                                 //  4,194,304  (65536 x 64)
    float* XR   = XH   + (size_t)MTOT * D_MODEL;      //  8,388,608  (65536 x 128)
    float* Z    = XR   + (size_t)MTOT * D_INNER;      //  8,388,608
    float* XC   = Z    + (size_t)MTOT * D_INNER;      //  8,388,608
    float* XDBL = XC   + (size_t)MTOT * D_INNER;      //  3,145,728  (65536 x 48)
    float* WP   = XDBL + (size_t)MTOT * NPAD;         //  6,144
    float* YG   = XR;                                 // alias: XR dead after conv
    float* Mb   = Z;                                  // alias: Z dead after scan
    float* XM   = Z + (size_t)MTOT * D_MODEL;         // alias: second half of Z
    float* POUT = XC;                                 // alias: XC dead after scan
    float* TMP  = XC + (size_t)RTOT2 * CHN;           // alias: upper part of XC

    // 0) zero-pad x_proj weights to N=48
    k_pad_xproj<<<(NPAD * D_INNER + 255) / 256, 256, 0, stream>>>(x_proj_w, WP);
    // 1) LayerNorm + split halves -> XH
    k_ln_split<<<(RTOT2 + 255) / 256, 256, 0, stream>>>(rgb, ir, ln_w, ln_b, XH);
    // 2) in_proj: xr half and z half (shared weights across the 4 instances)
    launch_gemm<EPI_NONE, false>(stream, XH, XH, 0, MTOT, D_INNER, D_MODEL, D_MODEL,
                                 in_proj_w, D_MODEL, XR, D_INNER,
                                 nullptr, nullptr, nullptr, 0, nullptr, nullptr, 0);
    launch_gemm<EPI_NONE, false>(stream, XH, XH, 0, MTOT, D_INNER, D_MODEL, D_MODEL,
                                 in_proj_w + (size_t)D_INNER * D_MODEL, D_MODEL, Z, D_INNER,
                                 nullptr, nullptr, nullptr, 0, nullptr, nullptr, 0);
    // 3) depthwise causal conv + SiLU -> XC
    k_conv_silu<<<(int)(((long)MTOT * D_INNER + 255) / 256), 256, 0, stream>>>(XR, conv_w, conv_b, XC);
    // 4) x_proj (N padded to 48) -> XDBL
    launch_gemm<EPI_NONE, false>(stream, XC, XC, 0, MTOT, NPAD, D_INNER, D_INNER,
                                 WP, D_INNER, XDBL, NPAD,
                                 nullptr, nullptr, nullptr, 0, nullptr, nullptr, 0);
    // 5) fused dt_proj + softplus + selective scan + D-skip + SiLU(z) gate -> YG
    k_scan<<<16, 128, 0, stream>>>(XDBL, XC, Z, dt_proj_w, dt_proj_b, A_log, D_skip, YG);
    // 6) out_proj + skip_scale * x_half -> Mb
    launch_gemm<EPI_SKIP, false>(stream, YG, YG, 0, MTOT, D_MODEL, D_INNER, D_INNER,
                                 out_proj_w, D_INNER, Mb, D_MODEL,
                                 nullptr, skip_scale, XH, D_MODEL, nullptr, nullptr, 0);
    // 7) LayerNorm over concat(m1,m2) -> XM
    k_ln2<<<(RTOT2 + 255) / 256, 256, 0, stream>>>(Mb, ln_w, ln_b, XM);
    // 8) proj + bias, scaled by per-branch weight -> POUT
    launch_gemm<EPI_BIAS_WSEL, false>(stream, XM, XM, 0, RTOT2, CHN, CHN, CHN,
                                      proj_w, CHN, POUT, CHN,
                                      proj_b, nullptr, nullptr, 0, weight_rgb, weight_ir, BL);
    // 9) convtout 1x1 over concatenated branch channels (split-A, K=256) -> TMP
    launch_gemm<EPI_BIAS, true>(stream, POUT, POUT + (size_t)BL * CHN, CHN, BL, CHN, 2 * CHN, CHN,
                                convtout_w, 2 * CHN, TMP, CHN,
                                convtout_b, nullptr, nullptr, 0, nullptr, nullptr, 0);
    // 10) (b,l,o) -> (b,o,h,w)
    k_transpose_out<<<(BATCH * CHN * HW + 255) / 256, 256, 0, stream>>>(TMP, out);
}